// LearningEngine_33062658245232
// MI455X (gfx1250) — compile-verified
//
#include <hip/hip_runtime.h>
#include <math.h>

// ---------------- problem constants ----------------
#define Hh     6
#define Nn     16
#define BATCH  32
#define Dd     512
#define Rr     64
#define Mm     80            // (H-1)*N
#define BD     16384         // BATCH*D
#define LVL    262144        // N*BATCH*D  (one T level)
#define DR     32768         // D*R

#define ETA_W   0.015f
#define LAM_C   0.65f
#define LAM_R   0.35f
#define THETA0  0.5f
#define SPEC_MIN 0.3f
#define SPEC_MAX 4.0f
#define W_MAX   45.254833995939045f     // sqrt(512)*2
#define DW_CLIP 5.0f
#define INV_SQRT_D 0.04419417382415922f // 1/sqrt(512)

typedef __attribute__((ext_vector_type(2))) float v2f;
typedef __attribute__((ext_vector_type(8))) float v8f;

// Block-wide sum for blockDim.x==256 (8 waves of 32). Every thread returns the total.
__device__ __forceinline__ float blk_sum(float v, float* red) {
    int t = threadIdx.x;
    #pragma unroll
    for (int off = 16; off; off >>= 1) v += __shfl_down(v, off, 32);
    __syncthreads();                       // protect red from previous use
    if ((t & 31) == 0) red[t >> 5] = v;
    __syncthreads();
    float s = red[0];
    #pragma unroll
    for (int w = 1; w < 8; ++w) s += red[w];
    return s;
}

// ------------------------------------------------------------------
// K1: one pass over V_edges. Per (h,i) block: V_in[b,d] = sum_j V[h,i,j,b,d]
//     and norms[h,i,j] = mean_b || V[h,i,j,b,:] ||
// ------------------------------------------------------------------
__global__ void k_vedge(const float* __restrict__ V, float* __restrict__ Vin,
                        float* __restrict__ norms) {
    int blk = blockIdx.x;                 // h*16+i
    int t   = threadIdx.x;
    const float* base = V + (size_t)blk * (size_t)(Nn * BD);
    float* vout = Vin + (size_t)blk * BD;
    __shared__ float red[8];
    __shared__ float snorm[Nn];
    if (t < Nn) snorm[t] = 0.f;
    __syncthreads();
    for (int b = 0; b < BATCH; ++b) {
        float v0 = 0.f, v1 = 0.f;
        for (int j = 0; j < Nn; ++j) {
            const float* p = base + ((size_t)(j * BATCH + b) * Dd);
            if (j < Nn - 1) __builtin_prefetch(p + (size_t)BATCH * Dd + t, 0, 0);
            float a0 = p[t], a1 = p[t + 256];
            v0 += a0; v1 += a1;
            float tot = blk_sum(a0 * a0 + a1 * a1, red);
            if (t == 0) snorm[j] += sqrtf(tot);
        }
        vout[b * Dd + t]       = v0;
        vout[b * Dd + t + 256] = v1;
    }
    __syncthreads();
    if (t < Nn) norms[blk * Nn + t] = snorm[t] * (1.f / (float)BATCH);
}

// ------------------------------------------------------------------
// K2: w_blame[h,a,:] = norms[h,a,:] / (sum + 1e-9)
// ------------------------------------------------------------------
__global__ void k_wblame(const float* __restrict__ norms, float* __restrict__ wb) {
    int t = threadIdx.x;
    if (t < Mm) {
        float s = 0.f;
        #pragma unroll
        for (int j = 0; j < Nn; ++j) s += norms[t * Nn + j];
        float inv = 1.f / (s + 1e-9f);
        #pragma unroll
        for (int j = 0; j < Nn; ++j) wb[t * Nn + j] = norms[t * Nn + j] * inv;
    }
}

// ------------------------------------------------------------------
// K3: per-node spectral norm via 20 power iterations of u <- A Bt B At u
// ------------------------------------------------------------------
__global__ void k_spectral(const float* __restrict__ A, const float* __restrict__ B,
                           float* __restrict__ scale) {
    int m = blockIdx.x, t = threadIdx.x;
    const float* A1 = A + (size_t)m * DR;
    const float* B1 = B + (size_t)m * DR;
    __shared__ float u[Dd];
    __shared__ float w[Dd];
    __shared__ float t64[Rr];
    __shared__ float pbuf[256];
    __shared__ float red[8];

    u[t] = INV_SQRT_D; u[t + 256] = INV_SQRT_D;
    __syncthreads();
    const int r = t & 63, part = t >> 6;     // 4 parts x 128 d

    for (int iter = 0; iter < 21; ++iter) {
        // t64 = A^T u
        { float s = 0.f;
          for (int d = part * 128; d < part * 128 + 128; ++d) s += A1[(size_t)d * Rr + r] * u[d];
          pbuf[t] = s; __syncthreads();
          if (t < Rr) t64[t] = pbuf[t] + pbuf[t + 64] + pbuf[t + 128] + pbuf[t + 192];
          __syncthreads(); }
        // w = B t64
        for (int d = t; d < Dd; d += 256) {
            float s = 0.f;
            #pragma unroll
            for (int k = 0; k < Rr; ++k) s += B1[(size_t)d * Rr + k] * t64[k];
            w[d] = s;
        }
        __syncthreads();
        if (iter == 20) break;               // final: spec = ||B (A^T u)||
        // t64 = B^T w
        { float s = 0.f;
          for (int d = part * 128; d < part * 128 + 128; ++d) s += B1[(size_t)d * Rr + r] * w[d];
          pbuf[t] = s; __syncthreads();
          if (t < Rr) t64[t] = pbuf[t] + pbuf[t + 64] + pbuf[t + 128] + pbuf[t + 192];
          __syncthreads(); }
        // w = A t64
        for (int d = t; d < Dd; d += 256) {
            float s = 0.f;
            #pragma unroll
            for (int k = 0; k < Rr; ++k) s += A1[(size_t)d * Rr + k] * t64[k];
            w[d] = s;
        }
        __syncthreads();
        float n2  = blk_sum(w[t] * w[t] + w[t + 256] * w[t + 256], red);
        float inv = 1.f / (sqrtf(n2) + 1e-12f);
        u[t] = w[t] * inv; u[t + 256] = w[t + 256] * inv;
        __syncthreads();
    }
    float sp2 = blk_sum(w[t] * w[t] + w[t + 256] * w[t + 256], red);
    if (t == 0) {
        float sp = sqrtf(sp2);
        scale[m] = fminf(fmaxf(sp, SPEC_MIN), SPEC_MAX);
    }
}

// ------------------------------------------------------------------
// K4: mse  (single block)
// ------------------------------------------------------------------
__global__ void k_mse(const float* __restrict__ yh, const float* __restrict__ ys,
                      float* __restrict__ out) {
    __shared__ float red[8];
    int t = threadIdx.x;
    float s = 0.f;
    for (int i = t; i < BD; i += 256) { float d = ys[i] - yh[i]; s += d * d; }
    float tot = blk_sum(s, red);
    if (t == 0) out[0] = tot / (float)BD;
}

// ------------------------------------------------------------------
// K5: T_all[5]: row i=0 gets T_v, rest zero
// ------------------------------------------------------------------
__global__ void k_ttop(const float* __restrict__ yh, const float* __restrict__ ys,
                       float* __restrict__ T5) {
    int idx = blockIdx.x * 256 + threadIdx.x;   // < 262144
    int i = idx >> 14;
    float val = 0.f;
    if (i == 0) { int rem = idx & (BD - 1); val = (ys[rem] - yh[rem]) * INV_SQRT_D; }
    T5[idx] = val;
}

// ------------------------------------------------------------------
// K6: tnorm of incoming T level + edge tensions for step h
// ------------------------------------------------------------------
__global__ void k_tnorm_edge(const float* __restrict__ Tsrc, const float* __restrict__ wb,
                             const float* __restrict__ scale, float* __restrict__ out_edge,
                             int h) {
    int j = blockIdx.x, t = threadIdx.x;
    __shared__ float red[8];
    __shared__ float bc;
    float acc = 0.f;
    for (int b = 0; b < BATCH; ++b) {
        float x0 = Tsrc[(size_t)j * BD + b * Dd + t];
        float x1 = Tsrc[(size_t)j * BD + b * Dd + t + 256];
        float tot = blk_sum(x0 * x0 + x1 * x1, red);
        acc += sqrtf(tot);                 // identical on every thread
    }
    if (t == 0) bc = acc * (1.f / (float)BATCH);
    __syncthreads();
    if (t < Nn) {
        int jj = h * Nn + j;
        out_edge[jj * Nn + t] = scale[jj] * wb[jj * Nn + t] * bc;
    }
}

// ------------------------------------------------------------------
// K7: blame mixing via V_WMMA_F32_16X16X4_F32.
//   Tdst[i, c] = sum_j Mmat[i,j] * Tsrc[j, c],  Mmat[i,j] = scale[h,j]*wb[h,j,i]
//   16x16 applied to 16x16384; one 16-column tile per wave, K split in 4 chunks of 4.
//   A layout (16x4 f32): lanes 0-15 -> M=lane, VGPR0=K0/VGPR1=K1; lanes16-31 -> K2/K3.
//   B layout (4x16 f32): VGPR0 -> rows K0 (lanes0-15) / K2 (lanes16-31); VGPR1 -> K1/K3.
//   D layout (16x16 f32): VGPR v -> row v (lanes0-15) / v+8 (lanes16-31), col = lane&15.
// ------------------------------------------------------------------
__global__ void k_mix(const float* __restrict__ Tsrc, float* __restrict__ Tdst,
                      const float* __restrict__ wb, const float* __restrict__ scale, int h) {
    int t    = threadIdx.x;
    int wave = (blockIdx.x * blockDim.x + t) >> 5;   // 0..1023
    int lane = t & 31;
    int n    = lane & 15;                            // A-row index == B/D column index
    int hi   = lane >> 4;
    int col0 = wave * 16;
    const float* wbh = wb + h * 256;
    const float* sch = scale + h * Nn;

    v8f acc = {0.f, 0.f, 0.f, 0.f, 0.f, 0.f, 0.f, 0.f};
    #pragma unroll
    for (int c = 0; c < 4; ++c) {
        int j0 = 4 * c + 2 * hi;
        v2f a, b;
        a.x = sch[j0]     * wbh[j0 * Nn + n];        // Mmat[n][j0]
        a.y = sch[j0 + 1] * wbh[(j0 + 1) * Nn + n];  // Mmat[n][j0+1]
        b.x = Tsrc[(size_t)j0 * BD + col0 + n];
        b.y = Tsrc[(size_t)(j0 + 1) * BD + col0 + n];
        acc = __builtin_amdgcn_wmma_f32_16x16x4_f32(false, a, false, b,
                                                    (short)0, acc, false, false);
    }
    #pragma unroll
    for (int v = 0; v < 8; ++v) {
        int row = v + 8 * hi;
        Tdst[(size_t)row * BD + col0 + n] = acc[v];
    }
}

// ------------------------------------------------------------------
// K8: per-node fused update (rank-1 dW => Adam on A and B)
// ------------------------------------------------------------------
__global__ void k_update(const float* __restrict__ A,  const float* __restrict__ B,
                         const float* __restrict__ mA, const float* __restrict__ vA,
                         const float* __restrict__ mB, const float* __restrict__ vB,
                         const float* __restrict__ Vin_all, const float* __restrict__ T_all,
                         float* __restrict__ A_out, float* __restrict__ B_out) {
    int m = blockIdx.x, t = threadIdx.x;
    int hH = m >> 4, iN = m & 15;
    const size_t off = (size_t)m * DR;
    const float* A1  = A + off;
    const float* B1  = B + off;
    const float* Vin = Vin_all + (size_t)m * BD;
    const float* Tl  = T_all + ((size_t)(hH + 1) * Nn + iN) * BD;   // T_upd[m]

    __shared__ float P[BATCH * Rr];     // 32x64
    __shared__ float u[Dd];
    __shared__ float vv[Dd];
    __shared__ float q[Rr];
    __shared__ float p2[Rr];
    __shared__ float pbuf[256];
    __shared__ float red[8];

    // ---- P = Vin @ B1  (32x64) ----
    for (int k = 0; k < 8; ++k) {
        int o = t + k * 256;
        int b = o >> 6, r = o & 63;
        float s = 0.f;
        for (int d = 0; d < Dd; ++d) s += Vin[b * Dd + d] * B1[(size_t)d * Rr + r];
        P[o] = s;
    }
    __syncthreads();

    // ---- Vw = P @ A1^T ; tanh stats; per-d means ----
    float s1v[2], s2v[2], tmv[2], vmv[2];
    float gsum = 0.f, vinsq = 0.f;
    for (int half = 0; half < 2; ++half) {
        int d = t + half * 256;
        float Ar[Rr];
        #pragma unroll
        for (int r = 0; r < Rr; ++r) Ar[r] = A1[(size_t)d * Rr + r];
        float s1 = 0.f, s2 = 0.f, tm = 0.f, vm = 0.f;
        for (int b = 0; b < BATCH; ++b) {
            float vw = 0.f;
            #pragma unroll
            for (int r = 0; r < Rr; ++r) vw += P[b * Rr + r] * Ar[r];
            float vo = tanhf(vw);
            float fp = 1.f - vo * vo;
            gsum += vo * vo;
            s1 += vo * fp;
            s2 += fp;
            float vi = Vin[b * Dd + d];
            vm += vi; vinsq += vi * vi;
            tm += Tl[b * Dd + d];
        }
        s1v[half] = s1; s2v[half] = s2; tmv[half] = tm; vmv[half] = vm;
    }
    float goodness = blk_sum(gsum, red) * (1.f / (float)BATCH);
    float theta    = THETA0 * (blk_sum(vinsq, red) * (1.f / (float)BATCH) + 1e-9f);
    float delta    = goodness - theta;

    const float invB = 1.f / (float)BATCH;
    #pragma unroll
    for (int half = 0; half < 2; ++half) {
        int d = t + half * 256;
        float Vg  = (s1v[half] * invB) * delta;       // Vout_grad[d]
        float fm  = s2v[half] * invB;                 // fprime.mean(0)[d]
        float tmn = tmv[half] * invB;                 // T_local.mean(0)[d]
        u[d]  = LAM_C * Vg + LAM_R * tmn * fm;
        vv[d] = vmv[half] * invB;                     // Vin_mean[d]
    }
    __syncthreads();

    // ---- Frobenius clip of rank-1 dW = u (x) vv ----
    float un2 = blk_sum(u[t] * u[t] + u[t + 256] * u[t + 256], red);
    float vn2 = blk_sum(vv[t] * vv[t] + vv[t + 256] * vv[t + 256], red);
    float gnorm = sqrtf(un2) * sqrtf(vn2);
    float cf = (gnorm > DW_CLIP) ? (DW_CLIP / (gnorm + 1e-12f)) : 1.f;
    u[t] *= cf; u[t + 256] *= cf;
    __syncthreads();

    // ---- q = vv @ B1 ; p2 = u @ A1  (both length 64) ----
    {
        int r = t & 63, part = t >> 6;
        float s = 0.f, s2 = 0.f;
        for (int d = part * 128; d < part * 128 + 128; ++d) {
            s  += vv[d] * B1[(size_t)d * Rr + r];
            s2 += u[d]  * A1[(size_t)d * Rr + r];
        }
        pbuf[t] = s; __syncthreads();
        if (t < Rr) q[t] = pbuf[t] + pbuf[t + 64] + pbuf[t + 128] + pbuf[t + 192];
        __syncthreads();
        pbuf[t] = s2; __syncthreads();
        if (t < Rr) p2[t] = pbuf[t] + pbuf[t + 64] + pbuf[t + 128] + pbuf[t + 192];
        __syncthreads();
    }

    // ---- Adam updates (gA = u (x) q, gB = vv (x) p2) + clamp ----
    for (int k = 0; k < 128; ++k) {
        int idx = t + k * 256;
        int d = idx >> 6, r = idx & 63;
        float g  = u[d] * q[r];
        float m1 = 0.9f * mA[off + idx] + 0.1f * g;
        float w1 = 0.999f * vA[off + idx] + 0.001f * g * g;
        float an = A1[idx] - ETA_W * (m1 * 10.f) / (sqrtf(w1 * 1000.f) + 1e-8f);
        A_out[off + idx] = fminf(fmaxf(an, -W_MAX), W_MAX);

        float g2 = vv[d] * p2[r];
        float m2 = 0.9f * mB[off + idx] + 0.1f * g2;
        float w2 = 0.999f * vB[off + idx] + 0.001f * g2 * g2;
        float bn = B1[idx] - ETA_W * (m2 * 10.f) / (sqrtf(w2 * 1000.f) + 1e-8f);
        B_out[off + idx] = fminf(fmaxf(bn, -W_MAX), W_MAX);
    }
}

// ------------------------------------------------------------------
extern "C" void kernel_launch(void* const* d_in, const int* in_sizes, int n_in,
                              void* d_out, int out_size, void* d_ws, size_t ws_size,
                              hipStream_t stream) {
    const float* Yh = (const float*)d_in[0];
    const float* Ys = (const float*)d_in[1];
    const float* V  = (const float*)d_in[2];
    const float* A  = (const float*)d_in[3];
    const float* Bw = (const float*)d_in[4];
    const float* mA = (const float*)d_in[5];
    const float* vA = (const float*)d_in[6];
    const float* mB = (const float*)d_in[7];
    const float* vB = (const float*)d_in[8];

    float* out      = (float*)d_out;
    float* out_mse  = out;
    float* out_edge = out + 1;
    float* out_A    = out + 1 + Hh * 0 + 5 * Nn * Nn;          // 1 + 1280
    float* out_B    = out_A + (size_t)(Hh - 1) * Nn * DR;      // + 2621440
    float* out_T    = out_B + (size_t)(Hh - 1) * Nn * DR;      // + 2621440

    float* ws       = (float*)d_ws;
    float* ws_Vin   = ws;                                      // 80*16384 = 1310720
    float* ws_norms = ws_Vin + (size_t)Mm * BD;                // 1280
    float* ws_wb    = ws_norms + Mm * Nn;                      // 1280
    float* ws_scale = ws_wb + Mm * Nn;                         // 80

    k_vedge   <<<Mm,   256, 0, stream>>>(V, ws_Vin, ws_norms);
    k_wblame  <<<1,    128, 0, stream>>>(ws_norms, ws_wb);
    k_spectral<<<Mm,   256, 0, stream>>>(A, Bw, ws_scale);
    k_mse     <<<1,    256, 0, stream>>>(Yh, Ys, out_mse);
    k_ttop    <<<1024, 256, 0, stream>>>(Yh, Ys, out_T + (size_t)5 * LVL);

    for (int h = 4; h >= 0; --h) {
        const float* Tsrc = out_T + (size_t)(h + 1) * LVL;
        float*       Tdst = out_T + (size_t)h * LVL;
        k_tnorm_edge<<<Nn,  256, 0, stream>>>(Tsrc, ws_wb, ws_scale, out_edge, h);
        k_mix       <<<128, 256, 0, stream>>>(Tsrc, Tdst, ws_wb, ws_scale, h);
    }

    k_update<<<Mm, 256, 0, stream>>>(A, Bw, mA, vA, mB, vB, ws_Vin, out_T, out_A, out_B);
}